// BaselineRNN_10874857193554
// MI455X (gfx1250) — compile-verified
//
#include <hip/hip_runtime.h>
#include <math.h>

typedef __attribute__((ext_vector_type(2))) float v2f;
typedef __attribute__((ext_vector_type(8))) float v8f;

namespace {
constexpr int kL = 4;   // layers
constexpr int kH = 8;   // hidden size
}

__device__ __forceinline__ float tanh_fast(float x) {
#if __has_builtin(__builtin_amdgcn_tanhf)
  return __builtin_amdgcn_tanhf(x);      // v_tanh_f32 on gfx1250
#else
  return tanhf(x);
#endif
}

// Latency-bound problem: only batch row B-1 reaches the output, so one wave32
// runs a systolic layer pipeline (layer l handles timestep k-l). Each layer's
// matvec (8x8 in layer 0, else 8x16) is a chain of V_WMMA_F32_16X16X4_F32; A rows 8-15 duplicate rows
// 0-7 and B columns are broadcast, so every lane holds the full 8-vector result
// in D and the recurrence needs zero cross-lane traffic.
extern "C" __global__ __launch_bounds__(32)
void rnn_wmma_pipeline(const float* __restrict__ x,
                       const float* __restrict__ h0,
                       const float* __restrict__ w_ih0,
                       const float* __restrict__ w_ih,
                       const float* __restrict__ w_hh,
                       const float* __restrict__ b_ih,
                       const float* __restrict__ b_hh,
                       const float* __restrict__ w_lin,
                       const float* __restrict__ b_lin,
                       float* __restrict__ out,
                       int T, int B)
{
  extern __shared__ float xs[];          // T + kL floats, zero-padded tail
  const int  lane   = (int)threadIdx.x;
  const int  row    = lane & 7;          // weight row (rows 8-15 duplicate 0-7)
  const bool hiHalf = (lane & 16) != 0;  // upper half-wave holds K+2 slices

  // Stage the single live batch row of x into LDS; pad so the steady-state
  // loop needs no bounds check.
  for (int t = lane; t < T; t += 32)
    xs[t] = x[(size_t)t * (size_t)B + (size_t)(B - 1)];
  if (lane < kL) xs[T + lane] = 0.0f;
  __syncthreads();

  // ---- loop-invariant operands ----
  // A layout (16x4 f32): VGPR v -> K = base + v + (hiHalf ? 2 : 0).
  v2f A0[2];                 // layer 0: W_hh only (x-term folded into C)
#pragma unroll
  for (int kk = 0; kk < 2; ++kk)
#pragma unroll
    for (int v = 0; v < 2; ++v)
      A0[kk][v] = w_hh[row * kH + 4 * kk + v + (hiHalf ? 2 : 0)];

  v2f A[kL - 1][4];          // layers 1..3: M = [W_hh | W_ih] (8x16)
#pragma unroll
  for (int l = 1; l < kL; ++l) {
#pragma unroll
    for (int kk = 0; kk < 4; ++kk) {
#pragma unroll
      for (int v = 0; v < 2; ++v) {
        const int K = 4 * kk + v + (hiHalf ? 2 : 0);
        A[l - 1][kk][v] = (K < kH)
            ? w_hh[(l * kH + row) * kH + K]
            : w_ih[((l - 1) * kH + row) * kH + (K - kH)];
      }
    }
  }

  v8f Cb[kL];                // fused bias, broadcast to all lanes
#pragma unroll
  for (int l = 0; l < kL; ++l)
#pragma unroll
    for (int m = 0; m < kH; ++m)
      Cb[l][m] = b_ih[l * kH + m] + b_hh[l * kH + m];

  float w0[kH];
#pragma unroll
  for (int m = 0; m < kH; ++m) w0[m] = w_ih0[m];

  float wl[kH];
#pragma unroll
  for (int m = 0; m < kH; ++m) wl[m] = w_lin[m];
  const float blin = b_lin[0];

  float hinit[kL][kH];
#pragma unroll
  for (int l = 0; l < kL; ++l)
#pragma unroll
    for (int m = 0; m < kH; ++m)
      hinit[l][m] = h0[((size_t)l * (size_t)B + (size_t)(B - 1)) * kH + m];

  // Pipeline state (lane-uniform): h = recurrent, g = input from layer below
  float h[kL][kH], g[kL][kH];
#pragma unroll
  for (int l = 0; l < kL; ++l)
#pragma unroll
    for (int m = 0; m < kH; ++m) { h[l][m] = 0.0f; g[l][m] = 0.0f; }

  const int kEnd = T + kL - 1;           // layer 3 finishes t = T-1 at k = T+2
  float xcur = xs[0];
  for (int k = 0; k < kEnd; ++k) {
    const float xnxt = xs[k + 1];        // padded: safe, zero past T

    // Inject initial hidden state exactly when layer k reaches t == 0.
    // Single uniform guard; not taken in steady state.
    if (__builtin_expect(k < kL, 0)) {
#pragma unroll
      for (int l = 0; l < kL; ++l) {
        if (k == l) {
#pragma unroll
          for (int m = 0; m < kH; ++m) h[l][m] = hinit[l][m];
        }
      }
    }

    v8f D[kL];

    // ---- layer 0: D = W_hh0 @ h0 + (bias + x_t * w_ih0), 2-deep chain ----
    {
      v8f c;
#pragma unroll
      for (int m = 0; m < kH; ++m) c[m] = fmaf(xcur, w0[m], Cb[0][m]);
      v2f b0, b1;
      b0[0] = hiHalf ? h[0][2] : h[0][0];
      b0[1] = hiHalf ? h[0][3] : h[0][1];
      b1[0] = hiHalf ? h[0][6] : h[0][4];
      b1[1] = hiHalf ? h[0][7] : h[0][5];
      c = __builtin_amdgcn_wmma_f32_16x16x4_f32(false, A0[0], false, b0, (short)0, c, false, false);
      c = __builtin_amdgcn_wmma_f32_16x16x4_f32(false, A0[1], false, b1, (short)0, c, false, false);
      D[0] = c;
    }

    // ---- layers 1..3: D = [W_hh|W_ih] @ [h;g] + bias, 4-deep chains ----
#pragma unroll
    for (int l = 1; l < kL; ++l) {
      v2f Bv[4];
#pragma unroll
      for (int kk = 0; kk < 4; ++kk) {
#pragma unroll
        for (int v = 0; v < 2; ++v) {
          const int jlo = 4 * kk + v;        // K index for lanes 0-15
          const int jhi = jlo + 2;           // K index for lanes 16-31
          const float slo = (jlo < kH) ? h[l][jlo] : g[l][jlo - kH];
          const float shi = (jhi < kH) ? h[l][jhi] : g[l][jhi - kH];
          Bv[kk][v] = hiHalf ? shi : slo;
        }
      }
      v8f d = Cb[l];
      d = __builtin_amdgcn_wmma_f32_16x16x4_f32(false, A[l - 1][0], false, Bv[0], (short)0, d, false, false);
      d = __builtin_amdgcn_wmma_f32_16x16x4_f32(false, A[l - 1][1], false, Bv[1], (short)0, d, false, false);
      d = __builtin_amdgcn_wmma_f32_16x16x4_f32(false, A[l - 1][2], false, Bv[2], (short)0, d, false, false);
      d = __builtin_amdgcn_wmma_f32_16x16x4_f32(false, A[l - 1][3], false, Bv[3], (short)0, d, false, false);
      D[l] = d;
    }

    // ---- tanh + systolic state update (no cross-lane traffic) ----
    float hn[kL][kH];
#pragma unroll
    for (int l = 0; l < kL; ++l)
#pragma unroll
      for (int m = 0; m < kH; ++m)
        hn[l][m] = tanh_fast(D[l][m]);

#pragma unroll
    for (int l = 0; l < kL; ++l) {
#pragma unroll
      for (int m = 0; m < kH; ++m) {
        h[l][m] = hn[l][m];
        if (l + 1 < kL) g[l + 1][m] = hn[l][m];
      }
    }

    // ---- linear head on layer-3 output ----
    if (k >= kL - 1) {
      float acc = blin;
#pragma unroll
      for (int m = 0; m < kH; ++m) acc = fmaf(hn[kL - 1][m], wl[m], acc);
      if (lane == 0) out[k - (kL - 1)] = acc;
    }

    xcur = xnxt;
  }
}

extern "C" void kernel_launch(void* const* d_in, const int* in_sizes, int n_in,
                              void* d_out, int out_size, void* d_ws, size_t ws_size,
                              hipStream_t stream) {
  (void)n_in; (void)d_ws; (void)ws_size; (void)out_size;
  const float* x     = (const float*)d_in[0];
  const float* h0    = (const float*)d_in[1];
  const float* w_ih0 = (const float*)d_in[2];
  const float* w_ih  = (const float*)d_in[3];
  const float* w_hh  = (const float*)d_in[4];
  const float* b_ih  = (const float*)d_in[5];
  const float* b_hh  = (const float*)d_in[6];
  const float* w_lin = (const float*)d_in[7];
  const float* b_lin = (const float*)d_in[8];
  float* out = (float*)d_out;

  const int B = in_sizes[1] / (kL * kH);   // initial_h is (L, B, H)
  const int T = in_sizes[0] / B;           // x is (T, B, 1)
  const size_t shmem = (size_t)(T + kL) * sizeof(float);

  rnn_wmma_pipeline<<<1, 32, shmem, stream>>>(x, h0, w_ih0, w_ih, w_hh,
                                              b_ih, b_hh, w_lin, b_lin,
                                              out, T, B);
}